// SO3Conv_29807073034808
// MI455X (gfx1250) — compile-verified
//
#include <hip/hip_runtime.h>
#include <hip/hip_bf16.h>

// ---------------------------------------------------------------------------
// SO(3) conv: psi = (D^T w)/sqrt(512) (scales folded), then per-l block GEMMs.
// Memory-bound (~220MB @ 23.3TB/s ~ 9.5us vs 48 GFLOP -> needs only ~5 TFLOP/s),
// so keep exact f32 via V_WMMA_F32_16X16X4_F32 and optimize data movement:
// double-buffered GLOBAL_LOAD_ASYNC_TO_LDS tile pipeline (ASYNCcnt) with
// k-contiguous A and kernel-2 B LDS layouts so those fragments are one ds_load_b64.
// ---------------------------------------------------------------------------

typedef __attribute__((ext_vector_type(2))) float v2f;
typedef __attribute__((ext_vector_type(8))) float v8f;
typedef __attribute__((ext_vector_type(2))) int   i2v;
typedef __attribute__((ext_vector_type(4))) int   i4v;

#define S_TOT 286
#define F_IN 128
#define F_OUT 128
#define NGRID 512

// ---- CDNA5 async global->LDS path (guarded; sync fallback keeps same shape) --
#if defined(__AMDGCN__) &&                                                   \
    __has_builtin(__builtin_amdgcn_global_load_async_to_lds_b32) &&          \
    __has_builtin(__builtin_amdgcn_global_load_async_to_lds_b64) &&          \
    __has_builtin(__builtin_amdgcn_global_load_async_to_lds_b128) &&         \
    __has_builtin(__builtin_amdgcn_s_wait_asynccnt)
#define USE_ASYNC 1
#else
#define USE_ASYNC 0
#endif

// Builtin param types (from clang diagnostics): src = AS(1) typed ptr,
// dst = AS(3) typed ptr; pointee is int / int2 / int4 per width.
typedef __attribute__((address_space(1))) int as1_i32;
typedef __attribute__((address_space(3))) int as3_i32;
typedef __attribute__((address_space(1))) i2v as1_i2;
typedef __attribute__((address_space(3))) i2v as3_i2;
typedef __attribute__((address_space(1))) i4v as1_i4;
typedef __attribute__((address_space(3))) i4v as3_i4;

// generic->AS casts via integers: global keeps the 64-bit address; generic LDS
// pointers carry the LDS byte offset in their low 32 bits.
#define G64(p) ((unsigned long long)(const void*)(p))
#define L32(p) ((unsigned int)(unsigned long long)(void*)(p))

__device__ __forceinline__ void cp_b32(float* lds_dst, const float* gsrc) {
#if USE_ASYNC
  __builtin_amdgcn_global_load_async_to_lds_b32((as1_i32*)G64(gsrc),
                                                (as3_i32*)L32(lds_dst), 0, 0);
#else
  *lds_dst = *gsrc;
#endif
}
__device__ __forceinline__ void cp_b64(float* lds_dst, const float* gsrc) {
#if USE_ASYNC
  __builtin_amdgcn_global_load_async_to_lds_b64((as1_i2*)G64(gsrc),
                                                (as3_i2*)L32(lds_dst), 0, 0);
#else
  lds_dst[0] = gsrc[0];
  lds_dst[1] = gsrc[1];
#endif
}
__device__ __forceinline__ void cp_b128(float* lds_dst, const float* gsrc) {
#if USE_ASYNC
  __builtin_amdgcn_global_load_async_to_lds_b128((as1_i4*)G64(gsrc),
                                                 (as3_i4*)L32(lds_dst), 0, 0);
#else
  *reinterpret_cast<float4*>(lds_dst) = *reinterpret_cast<const float4*>(gsrc);
#endif
}
template <int N>
__device__ __forceinline__ void async_wait() {
#if USE_ASYNC
  __builtin_amdgcn_s_wait_asynccnt(N);  // leave at most the N newest in flight
#endif
}

// ---------------------------------------------------------------------------
// Kernel 1: psi[p, i] = sum_k w[p, k] * D[k, i] * rsqrt(512*128*d(l(i)))
//   M = 16384 (p = x*F_OUT+y), K = 512, N = 286. Grid (256, 5), block 256.
// LDS: As[m][k] (k-contig, async b128 fill), Bs[k][n] (matches D row order,
// async b64 fill with clamped edge - garbage columns are never stored).
// Double-buffered: 3 async ops / thread / tile.
// ---------------------------------------------------------------------------
__global__ __launch_bounds__(256) void psi_proj_kernel(
    const float* __restrict__ w, const float* __restrict__ Dm,
    float* __restrict__ psi) {
  __shared__ float As[2][64][20];  // [buf][m][k] pad 20 -> b64 frags bank-clean
  __shared__ float Bs[2][16][72];  // [buf][k][n] pad 72 -> halves on disjoint banks

  const int tid  = threadIdx.x;
  const int lane = tid & 31;
  const int wave = tid >> 5;
  const int wm   = wave & 3;   // M sub-tile 16*wm
  const int wn   = wave >> 2;  // N sub-tile 32*wn
  const int half = lane >> 4;  // k-group (ISA 16x4 f32 frag: lanes>=16 hold k+2)
  const int lm   = lane & 15;

  const int m0 = blockIdx.x * 64;
  const int n0 = blockIdx.y * 64;

  const int am = tid >> 2;        // A: m row per thread
  const int ak = (tid & 3) * 4;   // A: 4 contiguous k's -> one b128
  const int bk = tid >> 4;        // B: k row per thread
  const int bn = (tid & 15) * 4;  // B: 4 n's -> two b64 chunks

  // Clamp B chunk starts so reads stay inside D; clamped chunks only land in
  // LDS columns whose outputs (n >= 286) are never stored.
  int c0 = n0 + bn;     c0 = (c0 > 284) ? 284 : c0;
  int c1 = n0 + bn + 2; c1 = (c1 > 284) ? 284 : c1;

  const float* wrow = w + (size_t)(m0 + am) * NGRID + ak;

  auto stage = [&](int k0, int b) {
    cp_b128(&As[b][am][ak], wrow + k0);
    const float* drow = Dm + (size_t)(k0 + bk) * S_TOT;
    cp_b64(&Bs[b][bk][bn], drow + c0);
    cp_b64(&Bs[b][bk][bn + 2], drow + c1);
  };

  v8f acc0 = {};
  v8f acc1 = {};

  stage(0, 0);
  const int T = NGRID / 16;  // 32 tiles
  for (int t = 0; t < T; ++t) {
    const int cur = t & 1;
    if (t + 1 < T) {
      stage((t + 1) * 16, cur ^ 1);
      async_wait<3>();  // previous tile's 3 ops done; next tile's stay in flight
    } else {
      async_wait<0>();
    }
    __syncthreads();

#pragma unroll
    for (int kk = 0; kk < 4; ++kk) {
      const int kb = kk * 4 + 2 * half;
      const v2f a = *reinterpret_cast<const v2f*>(&As[cur][wm * 16 + lm][kb]);
      v2f b0, b1;
      b0.x = Bs[cur][kb + 0][wn * 32 + lm];
      b0.y = Bs[cur][kb + 1][wn * 32 + lm];
      b1.x = Bs[cur][kb + 0][wn * 32 + 16 + lm];
      b1.y = Bs[cur][kb + 1][wn * 32 + 16 + lm];
      acc0 = __builtin_amdgcn_wmma_f32_16x16x4_f32(false, a, false, b0,
                                                   (short)0, acc0, false, false);
      acc1 = __builtin_amdgcn_wmma_f32_16x16x4_f32(false, a, false, b1,
                                                   (short)0, acc1, false, false);
    }
    __syncthreads();
  }

  // Store with fused normalization 1/sqrt(NGRID * F_IN * d(l(i)))
#pragma unroll
  for (int sub = 0; sub < 2; ++sub) {
    const int n = n0 + wn * 32 + sub * 16 + lm;
    if (n < S_TOT) {
      const int dl = (n < 1) ? 1 : (n < 10) ? 3 : (n < 35) ? 5
                   : (n < 84) ? 7 : (n < 165) ? 9 : 11;
      const float sc = rsqrtf((float)NGRID * (float)F_IN * (float)dl);
      const v8f& acc = sub ? acc1 : acc0;
#pragma unroll
      for (int r = 0; r < 8; ++r) {
        const int m = m0 + wm * 16 + r + half * 8;  // C layout: vgpr r -> M=r / r+8
        psi[(size_t)m * S_TOT + n] = acc[r] * sc;
      }
    }
  }
}

// ---------------------------------------------------------------------------
// Kernel 2 (per l): fixed (l, i=blockIdx.z):
//   out[b, q] = sum_k x[b, k/d, (k%d)*d + i + off] * psi[k/d, q/d, (k%d)*d + q%d + off]
// M=512, N=K=128d (all tiles exact, no guards). Gather tiles element-wise via
// async b32 into k-contiguous LDS; incremental div/mod update (no per-tile idiv).
// 8 async ops / thread / tile, double-buffered.
// ---------------------------------------------------------------------------
__global__ __launch_bounds__(256) void so3_mix_kernel(
    const float* __restrict__ x, const float* __restrict__ psi,
    float* __restrict__ out, int d, int off) {
  __shared__ float As[2][64][20];  // [buf][m=b][k]
  __shared__ float Bs[2][64][20];  // [buf][n=q][k]

  const int tid  = threadIdx.x;
  const int lane = tid & 31;
  const int wave = tid >> 5;
  const int wm   = wave & 3;
  const int wn   = wave >> 2;
  const int half = lane >> 4;
  const int lm   = lane & 15;

  const int m0 = blockIdx.x * 64;  // batch base
  const int n0 = blockIdx.y * 64;  // q base
  const int i  = blockIdx.z;       // irrep dim index

  const int am = tid >> 2;
  const int ak = (tid & 3) * 4;
  const int bk = tid >> 4;
  const int bn = (tid & 15) * 4;

  // Incremental (k / d, k % d) state; k advances by 16 per tile.
  const int q16 = 16 / d;
  const int r16 = 16 - q16 * d;
  int xcA[4], uA[4];
#pragma unroll
  for (int j = 0; j < 4; ++j) {
    const int k = ak + j;
    xcA[j] = k / d;
    uA[j]  = k - xcA[j] * d;
  }
  int xcB = bk / d;
  int uB  = bk - xcB * d;

  // B's n-side decomposition is tile-invariant: hoist it.
  int yB[4], vB[4];
#pragma unroll
  for (int j = 0; j < 4; ++j) {
    const int q = n0 + bn + j;
    yB[j] = q / d;
    vB[j] = q - yB[j] * d;
  }

  const float* xrow = x + (size_t)(m0 + am) * (F_IN * S_TOT) + off + i;

  auto stage = [&](int b) {
#pragma unroll
    for (int j = 0; j < 4; ++j)
      cp_b32(&As[b][am][ak + j], xrow + (size_t)xcA[j] * S_TOT + uA[j] * d);
    const float* prow = psi + (size_t)xcB * (F_OUT * S_TOT) + off + uB * d;
#pragma unroll
    for (int j = 0; j < 4; ++j)
      cp_b32(&Bs[b][bn + j][bk], prow + (size_t)yB[j] * S_TOT + vB[j]);
    // advance div/mod state to the next k-tile (k += 16)
#pragma unroll
    for (int j = 0; j < 4; ++j) {
      uA[j] += r16;
      const int c = (uA[j] >= d) ? 1 : 0;
      uA[j] -= c ? d : 0;
      xcA[j] += q16 + c;
    }
    uB += r16;
    const int cB = (uB >= d) ? 1 : 0;
    uB -= cB ? d : 0;
    xcB += q16 + cB;
  };

  v8f acc0 = {};
  v8f acc1 = {};

  const int T = (F_IN * d) / 16;  // 8*d tiles
  stage(0);
  for (int t = 0; t < T; ++t) {
    const int cur = t & 1;
    if (t + 1 < T) {
      stage(cur ^ 1);
      async_wait<8>();
    } else {
      async_wait<0>();
    }
    __syncthreads();

#pragma unroll
    for (int kk = 0; kk < 4; ++kk) {
      const int kb = kk * 4 + 2 * half;
      const v2f a  = *reinterpret_cast<const v2f*>(&As[cur][wm * 16 + lm][kb]);
      const v2f b0 = *reinterpret_cast<const v2f*>(&Bs[cur][wn * 32 + lm][kb]);
      const v2f b1 = *reinterpret_cast<const v2f*>(&Bs[cur][wn * 32 + 16 + lm][kb]);
      acc0 = __builtin_amdgcn_wmma_f32_16x16x4_f32(false, a, false, b0,
                                                   (short)0, acc0, false, false);
      acc1 = __builtin_amdgcn_wmma_f32_16x16x4_f32(false, a, false, b1,
                                                   (short)0, acc1, false, false);
    }
    __syncthreads();
  }

  // Scatter C into out[b, y, v*d + i + off]; scale already folded into psi.
#pragma unroll
  for (int sub = 0; sub < 2; ++sub) {
    const int q = n0 + wn * 32 + sub * 16 + lm;
    const int y = q / d;
    const int v = q - y * d;
    const size_t col = (size_t)y * S_TOT + off + v * d + i;
    const v8f& acc = sub ? acc1 : acc0;
#pragma unroll
    for (int r = 0; r < 8; ++r) {
      const int m = m0 + wm * 16 + r + half * 8;
      out[(size_t)m * (F_OUT * S_TOT) + col] = acc[r];
    }
  }
}

extern "C" void kernel_launch(void* const* d_in, const int* in_sizes, int n_in,
                              void* d_out, int out_size, void* d_ws, size_t ws_size,
                              hipStream_t stream) {
  const float* x  = (const float*)d_in[0];  // (512, 128, 286)
  const float* w  = (const float*)d_in[1];  // (128, 128, 512)
  const float* Dm = (const float*)d_in[2];  // (512, 286)
  float* out = (float*)d_out;               // (512, 128, 286)
  float* psi = (float*)d_ws;                // (128*128, 286) = 17.9 MB scratch

  (void)in_sizes; (void)n_in; (void)out_size; (void)ws_size;

  psi_proj_kernel<<<dim3(256, 5, 1), dim3(256, 1, 1), 0, stream>>>(w, Dm, psi);

  const int ds[6]   = {1, 3, 5, 7, 9, 11};
  const int offs[6] = {0, 1, 10, 35, 84, 165};
  for (int l = 0; l < 6; ++l) {
    const int d = ds[l];
    so3_mix_kernel<<<dim3(8, 2 * d, d), dim3(256, 1, 1), 0, stream>>>(
        x, psi, out, d, offs[l]);
  }
}